// FastAffine_89532888252550
// MI455X (gfx1250) — compile-verified
//
#include <hip/hip_runtime.h>
#include <hip/hip_bf16.h>

// ---------------------------------------------------------------------------
// FastAffine whitening for MI455X (gfx1250, wave32, WMMA + Tensor Data Mover).
//   C = A^T A + eps I ; eigh(C) ; R = C^{-1/2}, Rinv = C^{1/2}
//   out = [ A @ R , G @ Rinv ]
// GEMMs: v_wmma_f32_16x16x4_f32.  Global->LDS staging: tensor_load_to_lds
// (TDM) with LDS padding done by the DMA engine, double-buffered per block.
// ---------------------------------------------------------------------------

typedef __attribute__((ext_vector_type(2))) float    v2f;
typedef __attribute__((ext_vector_type(8))) float    v8f;
typedef __attribute__((ext_vector_type(4))) unsigned v4u;
typedef __attribute__((ext_vector_type(4))) int      v4i;
typedef __attribute__((ext_vector_type(8))) int      v8i;

#define EPSV 1e-6f
#define GRAM_BLOCKS 512
#define MM_BLOCKS 2048
#define LSTR 68                 // padded LDS row stride (floats): conflict-free
#define TILE_F (128 * LSTR)     // floats per 128-row LDS tile buffer
// ws layout (floats): [0 .. 512*4096) partial grams | C (4096) | R (4096) | Rinv (4096)

__device__ __forceinline__ v8f wmma_f32_4(v2f a, v2f b, v8f c) {
    // (neg_a, A, neg_b, B, c_mod, C, reuse_a, reuse_b)
    return __builtin_amdgcn_wmma_f32_16x16x4_f32(false, a, false, b, (short)0, c,
                                                 false, false);
}

// ---------------------------------------------------------------------------
// TDM: copy one 16x64 f32 tile (contiguous 4KB at gsrc) into LDS at byte
// offset lds_off, inserting 4 pad DWORDs after every 64 data DWORDs so rows
// land at stride LSTR=68 floats.  One descriptor per wave; TENSORcnt-tracked.
// D# group1 constants: data_size=4B | pad_enable | pad_interval=64DW(code 5)
//                      | pad_amount=4DW(code 3);  tensor_dim0=tile_dim0=64;
//                      tensor_dim1=tile_dim1=16;  tensor_dim0_stride=64.
// 6-arg toolchain form: (g0 v4u, g1 v8i, g2 v4i, g3 v4i, g4 v8i, cpol).
// ---------------------------------------------------------------------------
__device__ __forceinline__ void tdm_load_16x64(const float* gsrc, unsigned lds_off) {
    unsigned long long ga = (unsigned long long)(size_t)gsrc;
    unsigned lo = __builtin_amdgcn_readfirstlane((unsigned)ga);
    unsigned hi = __builtin_amdgcn_readfirstlane((unsigned)(ga >> 32));
    unsigned lw = __builtin_amdgcn_readfirstlane(lds_off);
    v4u g0 = { 1u,                                   // count=1, user mode
               lw,                                   // lds_addr (bytes)
               lo,                                   // global_addr[31:0]
               (hi & 0x01ffffffu) | 0x80000000u };   // global_addr[56:32] | type=2
    v8i g1 = { 0x07520000,          // data_size=2(4B), pad_en, intvl=5, amt=3
               0x00400000,          // tensor_dim0 = 64  (bits 79:48 low part)
               0x00100000,          // tensor_dim1 = 16  (bits 111:80 low part)
               0x00400000,          // tile_dim0  = 64  (bits 127:112)
               16,                  // tile_dim1  = 16  (bits 143:128)
               64,                  // tensor_dim0_stride = 64 (bits 207:160)
               0, 0 };
    v4i gz4 = { 0, 0, 0, 0 };       // 2D tensor: groups 2/3 unused
    v8i gz8 = { 0, 0, 0, 0, 0, 0, 0, 0 };
    __builtin_amdgcn_tensor_load_to_lds(g0, g1, gz4, gz4, gz8, 0);
}

// WMMA accumulate for the Gram kernel: two 16x16 C-tiles per wave.
__device__ __forceinline__ void gram_accum(const float* tp, int half, int l16,
                                           int mi0, int nj0, int mi1, int nj1,
                                           v8f& acc0, v8f& acc1) {
    #pragma unroll 4
    for (int kk = 0; kk < 128; kk += 4) {
        const int k0 = kk + 2 * half;
        v2f a0, b0, a1, b1;
        a0.x = tp[k0 * LSTR + mi0 + l16]; a0.y = tp[(k0 + 1) * LSTR + mi0 + l16];
        b0.x = tp[k0 * LSTR + nj0 + l16]; b0.y = tp[(k0 + 1) * LSTR + nj0 + l16];
        acc0 = wmma_f32_4(a0, b0, acc0);
        a1.x = tp[k0 * LSTR + mi1 + l16]; a1.y = tp[(k0 + 1) * LSTR + mi1 + l16];
        b1.x = tp[k0 * LSTR + nj1 + l16]; b1.y = tp[(k0 + 1) * LSTR + nj1 + l16];
        acc1 = wmma_f32_4(a1, b1, acc1);
    }
}

// WMMA 16x64 output strip for the matmul kernel.
__device__ __forceinline__ void mm_strip(const float* tp, const float* tm,
                                         float* __restrict__ Out,
                                         long long rowbase, long long nrows,
                                         int rbase, int half, int l16) {
    v8f acc0 = {}, acc1 = {}, acc2 = {}, acc3 = {};
    #pragma unroll
    for (int kk = 0; kk < 64; kk += 4) {
        const int k0 = kk + 2 * half;
        v2f a, b0, b1, b2, b3;
        a.x  = tp[(rbase + l16) * LSTR + k0]; a.y = tp[(rbase + l16) * LSTR + k0 + 1];
        b0.x = tm[k0 * LSTR +  0 + l16];      b0.y = tm[(k0 + 1) * LSTR +  0 + l16];
        b1.x = tm[k0 * LSTR + 16 + l16];      b1.y = tm[(k0 + 1) * LSTR + 16 + l16];
        b2.x = tm[k0 * LSTR + 32 + l16];      b2.y = tm[(k0 + 1) * LSTR + 32 + l16];
        b3.x = tm[k0 * LSTR + 48 + l16];      b3.y = tm[(k0 + 1) * LSTR + 48 + l16];
        acc0 = wmma_f32_4(a, b0, acc0);
        acc1 = wmma_f32_4(a, b1, acc1);
        acc2 = wmma_f32_4(a, b2, acc2);
        acc3 = wmma_f32_4(a, b3, acc3);
    }
    #pragma unroll
    for (int r = 0; r < 8; ++r) {
        const long long grow = rowbase + rbase + r + 8 * half;
        if (grow < nrows) {
            float* o = Out + grow * 64 + l16;
            o[ 0] = acc0[r];
            o[16] = acc1[r];
            o[32] = acc2[r];
            o[48] = acc3[r];
        }
    }
}

// guarded synchronous tile load (tail chunks only)
__device__ __forceinline__ void sync_load_tile(const float* __restrict__ In,
                                               long long row0, long long nrows,
                                               float* tp, int tid) {
    for (int i = tid; i < 128 * 16; i += 256) {
        const int r = i >> 4, cc = (i & 15) * 4;
        float4 v = {0.f, 0.f, 0.f, 0.f};
        const long long gr = row0 + r;
        if (gr < nrows) v = *(const float4*)(In + gr * 64 + cc);
        tp[r * LSTR + cc + 0] = v.x; tp[r * LSTR + cc + 1] = v.y;
        tp[r * LSTR + cc + 2] = v.z; tp[r * LSTR + cc + 3] = v.w;
    }
}

// ---------------------------------------------------------------------------
// Kernel 1: partial Gram.  Persistent blocks, TDM double-buffered tiles.
// ---------------------------------------------------------------------------
__global__ __launch_bounds__(256) void gram_partial(const float* __restrict__ A,
                                                    long long nrows,
                                                    float* __restrict__ partial) {
    __shared__ float smem[2 * TILE_F];          // 69.6 KB, two 128x68 buffers
    const int tid  = threadIdx.x;
    const int wave = tid >> 5;
    const int lane = tid & 31;
    const int half = lane >> 4;
    const int l16  = lane & 15;

    const int t0  = wave * 2, t1 = wave * 2 + 1;
    const int mi0 = (t0 >> 2) * 16, nj0 = (t0 & 3) * 16;
    const int mi1 = (t1 >> 2) * 16, nj1 = (t1 & 3) * 16;

    v8f acc0 = {}; v8f acc1 = {};

    const long long fc = nrows >> 7;            // full 128-row chunks
    long long c = blockIdx.x;
    int buf = 0;
    if (c < fc)
        tdm_load_16x64(A + (c * 128 + wave * 16) * 64,
                       (unsigned)((wave * 16 * LSTR) * 4));
    for (; c < fc; c += GRAM_BLOCKS) {
        const long long cn = c + GRAM_BLOCKS;
        if (cn < fc) {
            tdm_load_16x64(A + (cn * 128 + wave * 16) * 64,
                           (unsigned)((((buf ^ 1) * 128 + wave * 16) * LSTR) * 4));
            __builtin_amdgcn_s_wait_tensorcnt(1);   // older buffer complete
        } else {
            __builtin_amdgcn_s_wait_tensorcnt(0);
        }
        __syncthreads();
        gram_accum(smem + buf * TILE_F, half, l16, mi0, nj0, mi1, nj1, acc0, acc1);
        __syncthreads();
        buf ^= 1;
    }

    // tail chunk (< 128 rows): one designated block, guarded sync path
    if ((nrows & 127) && (long long)blockIdx.x == (fc % GRAM_BLOCKS)) {
        sync_load_tile(A, fc * 128, nrows, smem, tid);
        __syncthreads();
        gram_accum(smem, half, l16, mi0, nj0, mi1, nj1, acc0, acc1);
    }

    float* P = partial + (long long)blockIdx.x * 4096;
    #pragma unroll
    for (int r = 0; r < 8; ++r) {
        const int rowc = r + 8 * half;
        P[(mi0 + rowc) * 64 + nj0 + l16] = acc0[r];
        P[(mi1 + rowc) * 64 + nj1 + l16] = acc1[r];
    }
}

// ---------------------------------------------------------------------------
// Kernel 2: deterministic reduction of the partial grams.
// ---------------------------------------------------------------------------
__global__ __launch_bounds__(256) void reduce_gram(const float* __restrict__ partial,
                                                   float* __restrict__ C) {
    const int e = blockIdx.x * 256 + threadIdx.x;   // 16 x 256 = 4096
    float s = 0.f;
    for (int b = 0; b < GRAM_BLOCKS; ++b) s += partial[b * 4096 + e];
    C[e] = s;
}

// ---------------------------------------------------------------------------
// Kernel 3: 64x64 symmetric eigensolver (parallel cyclic Jacobi, tournament
// ordering) + whitening matrices.  One block, 64 threads.
// ---------------------------------------------------------------------------
__global__ __launch_bounds__(64) void eigh_whiten(const float* __restrict__ Cin,
                                                  float* __restrict__ R,
                                                  float* __restrict__ Rinv) {
    __shared__ float C[64][65];
    __shared__ float V[64][65];
    __shared__ int   perm[64];
    __shared__ float cs[32], sn[32];
    __shared__ int   pp[32], qq[32];
    __shared__ float w1[64], w2[64];

    const int tid = threadIdx.x;
    for (int j = 0; j < 64; ++j) {
        C[tid][j] = Cin[tid * 64 + j] + (j == tid ? EPSV : 0.f);
        V[tid][j] = (j == tid) ? 1.f : 0.f;
    }
    perm[tid] = tid;
    __syncthreads();

    for (int sweep = 0; sweep < 15; ++sweep) {
        for (int round = 0; round < 63; ++round) {
            if (tid < 32) {
                int p = perm[tid], q = perm[63 - tid];
                if (p > q) { int t = p; p = q; q = t; }
                pp[tid] = p; qq[tid] = q;
                const float apq = C[p][q];
                float c = 1.f, s = 0.f;
                if (fabsf(apq) > 1e-12f) {
                    const float tau = (C[q][q] - C[p][p]) / (2.f * apq);
                    const float t = copysignf(1.f, tau) /
                                    (fabsf(tau) + sqrtf(1.f + tau * tau));
                    c = 1.f / sqrtf(1.f + t * t);
                    s = t * c;
                }
                cs[tid] = c; sn[tid] = s;
            }
            __syncthreads();
            for (int i = 0; i < 32; ++i) {          // J^T * C (rows), disjoint pairs
                const int p = pp[i], q = qq[i];
                const float c = cs[i], s = sn[i];
                const float cp = C[p][tid], cq = C[q][tid];
                C[p][tid] = c * cp - s * cq;
                C[q][tid] = s * cp + c * cq;
            }
            __syncthreads();
            for (int i = 0; i < 32; ++i) {          // C * J (cols) and V = V * J
                const int p = pp[i], q = qq[i];
                const float c = cs[i], s = sn[i];
                const float cp = C[tid][p], cq = C[tid][q];
                C[tid][p] = c * cp - s * cq;
                C[tid][q] = s * cp + c * cq;
                const float vp = V[tid][p], vq = V[tid][q];
                V[tid][p] = c * vp - s * vq;
                V[tid][q] = s * vp + c * vq;
            }
            __syncthreads();
            if (tid == 0) {                         // rotate tournament
                const int last = perm[63];
                for (int i = 63; i > 1; --i) perm[i] = perm[i - 1];
                perm[1] = last;
            }
            __syncthreads();
        }
    }

    const float cl = fmaxf(C[tid][tid], EPSV);
    w1[tid] = 1.f / sqrtf(cl);
    w2[tid] = sqrtf(cl);
    __syncthreads();

    for (int j = 0; j < 64; ++j) {
        float r = 0.f, ri = 0.f;
        for (int k = 0; k < 64; ++k) {
            const float prod = V[tid][k] * V[j][k];
            r  += prod * w1[k];
            ri += prod * w2[k];
        }
        R[tid * 64 + j]    = r;
        Rinv[tid * 64 + j] = ri;
    }
}

// ---------------------------------------------------------------------------
// Kernel 4: Out = In @ Mx (nrows x 64 @ 64 x 64) via WMMA.  Persistent
// blocks, TDM double-buffered input tiles, Mx cached in LDS.
// ---------------------------------------------------------------------------
__global__ __launch_bounds__(256) void matmul64(const float* __restrict__ In,
                                                const float* __restrict__ Mx,
                                                float* __restrict__ Out,
                                                long long nrows) {
    __shared__ float smem[2 * TILE_F + 64 * LSTR];  // 87 KB
    float* tm = smem + 2 * TILE_F;
    const int tid  = threadIdx.x;
    const int wave = tid >> 5;
    const int lane = tid & 31;
    const int half = lane >> 4;
    const int l16  = lane & 15;
    const int rbase = wave * 16;

    for (int i = tid; i < 64 * 16; i += 256) {      // stage Mx (hot in L2)
        const int r = i >> 4, cc = (i & 15) * 4;
        const float4 v = *(const float4*)(Mx + r * 64 + cc);
        tm[r * LSTR + cc + 0] = v.x; tm[r * LSTR + cc + 1] = v.y;
        tm[r * LSTR + cc + 2] = v.z; tm[r * LSTR + cc + 3] = v.w;
    }

    const long long fc = nrows >> 7;
    long long c = blockIdx.x;
    int buf = 0;
    if (c < fc)
        tdm_load_16x64(In + (c * 128 + wave * 16) * 64,
                       (unsigned)((wave * 16 * LSTR) * 4));
    for (; c < fc; c += gridDim.x) {
        const long long cn = c + gridDim.x;
        if (cn < fc) {
            tdm_load_16x64(In + (cn * 128 + wave * 16) * 64,
                           (unsigned)((((buf ^ 1) * 128 + wave * 16) * LSTR) * 4));
            __builtin_amdgcn_s_wait_tensorcnt(1);
        } else {
            __builtin_amdgcn_s_wait_tensorcnt(0);
        }
        __syncthreads();                            // also publishes tm (1st iter)
        mm_strip(smem + buf * TILE_F, tm, Out, c * 128, nrows, rbase, half, l16);
        __syncthreads();
        buf ^= 1;
    }

    // tail chunk (< 128 rows): designated block, guarded sync path
    if ((nrows & 127) && (long long)blockIdx.x == (fc % gridDim.x)) {
        sync_load_tile(In, fc * 128, nrows, smem, tid);
        __syncthreads();
        mm_strip(smem, tm, Out, fc * 128, nrows, rbase, half, l16);
    }
}

// ---------------------------------------------------------------------------
extern "C" void kernel_launch(void* const* d_in, const int* in_sizes, int n_in,
                              void* d_out, int out_size, void* d_ws, size_t ws_size,
                              hipStream_t stream) {
    const float* A = (const float*)d_in[0];
    const float* G = (const float*)d_in[1];
    const long long nA = (long long)in_sizes[0] / 64;   // 1,000,000
    const long long nG = (long long)in_sizes[1] / 64;   //   200,000
    float* out = (float*)d_out;

    float* ws      = (float*)d_ws;
    float* partial = ws;                                  // 512*4096 floats (8 MB)
    float* C       = ws + (long long)GRAM_BLOCKS * 4096;
    float* R       = C + 4096;
    float* Rinv    = R + 4096;

    gram_partial<<<GRAM_BLOCKS, 256, 0, stream>>>(A, nA, partial);
    reduce_gram<<<16, 256, 0, stream>>>(partial, C);
    eigh_whiten<<<1, 64, 0, stream>>>(C, R, Rinv);

    matmul64<<<MM_BLOCKS, 256, 0, stream>>>(A, R, out, nA);
    matmul64<<<MM_BLOCKS, 256, 0, stream>>>(G, Rinv, out + nA * 64, nG);
}